// ProteinFeatures_75350906241675
// MI455X (gfx1250) — compile-verified
//
#include <hip/hip_runtime.h>

#define LRES  8192
#define KNB   48
#define NODEF 128
#define EDGEF 128
#define CIN   416   // 128 + 16 + 16 + 128 + 128

typedef float v2f __attribute__((ext_vector_type(2)));
typedef float v8f __attribute__((ext_vector_type(8)));
typedef int   v4i __attribute__((ext_vector_type(4)));

#if __has_builtin(__builtin_amdgcn_global_load_async_to_lds_b128) && \
    __has_builtin(__builtin_amdgcn_s_wait_asynccnt)
#define USE_ASYNC_GATHER 1
typedef __attribute__((address_space(3))) v4i* lds_v4i_p;
#define AS3V4I(p) \
  ((lds_v4i_p)(__attribute__((address_space(3))) void*)(void*)(p))
#endif

__device__ __forceinline__ v8f wmma4(v2f a, v2f b, v8f c) {
  // D = A(16x4 f32) * B(4x16 f32) + C(16x16 f32)
  return __builtin_amdgcn_wmma_f32_16x16x4_f32(false, a, false, b, (short)0, c,
                                               false, false);
}

__device__ __forceinline__ v8f splat8(float x) {
  v8f r;
#pragma unroll
  for (int i = 0; i < 8; ++i) r[i] = x;
  return r;
}

// ---------------------------------------------------------------------------
// Kernel 1: pack We[128][416] and Wproj[128][128] into WMMA B-fragment order.
// B[k][n] = W[n][k]. Fragment (ntile, cs, lane): lanes 0-15 -> K = 4cs..4cs+1,
// lanes 16-31 -> K = 4cs+2..4cs+3; N = ntile*16 + (lane&15). Stored as float2
// so edge_kernel does one coalesced b64 load per fragment.
// ---------------------------------------------------------------------------
__global__ __launch_bounds__(256) void pack_kernel(
    const float* __restrict__ We, const float* __restrict__ Wproj,
    v2f* __restrict__ WeP, v2f* __restrict__ WprojP) {
  int t = blockIdx.x * blockDim.x + threadIdx.x;
  const int NWE = 8 * 104 * 32;
  if (t < NWE) {
    int lane = t & 31;
    int cs = (t >> 5) % 104;
    int nt = t / (104 * 32);
    int n = nt * 16 + (lane & 15);
    int c = 4 * cs + 2 * (lane >> 4);
    v2f v;
    v.x = We[n * CIN + c];
    v.y = We[n * CIN + c + 1];
    WeP[t] = v;
  } else if (t < NWE + 8 * 32 * 32) {
    int u = t - NWE;
    int lane = u & 31;
    int cs = (u >> 5) & 31;
    int nt = u / (32 * 32);
    int n = nt * 16 + (lane & 15);
    int c = 4 * cs + 2 * (lane >> 4);
    v2f v;
    v.x = Wproj[n * EDGEF + c];
    v.y = Wproj[n * EDGEF + c + 1];
    WprojP[u] = v;
  }
}

// ---------------------------------------------------------------------------
// Kernel 2: per-residue k-NN. One 256-thread block per residue. Each thread
// owns 32 candidates in registers with an exclusion mask and a CACHED local
// (min,idx): after extracting a winner only the owning thread rescans.
// Cross-thread reduction = wave32 shfl_xor butterfly (barrier-free) + one
// 8-way scalar pass -> 2 block barriers per iteration instead of 8.
// Ties break to lowest index, matching jax.lax.top_k.
// ---------------------------------------------------------------------------
__global__ __launch_bounds__(256) void topk_kernel(const float* __restrict__ X,
                                                   int* __restrict__ idx_ws,
                                                   float* __restrict__ dk_ws) {
  const int i = blockIdx.x;
  const int t = threadIdx.x;
  const int w = t >> 5;
  const int lane = t & 31;
  __shared__ float swd[9];
  __shared__ int swi[9];

  const float xi0 = X[3 * i + 0], xi1 = X[3 * i + 1], xi2 = X[3 * i + 2];
  const int base = t * 32;
  float dl[32];
#pragma unroll 4
  for (int s = 0; s < 32; ++s) {
    int j = base + s;
    float dx = X[3 * j + 0] - xi0;
    float dy = X[3 * j + 1] - xi1;
    float dz = X[3 * j + 2] - xi2;
    dl[s] = sqrtf(dx * dx + dy * dy + dz * dz + 1e-6f);
  }
  unsigned int mask = 0u;
  float bd = 3.4e38f;
  int bi = 0x7fffffff;
#pragma unroll
  for (int s = 0; s < 32; ++s) {
    if (dl[s] < bd) { bd = dl[s]; bi = base + s; }
  }

  for (int it = 0; it < KNB; ++it) {
    // wave32 butterfly min-reduce of (d, idx)
    float rdv = bd;
    int riv = bi;
#pragma unroll
    for (int m = 16; m >= 1; m >>= 1) {
      float od = __shfl_xor(rdv, m, 32);
      int oi = __shfl_xor(riv, m, 32);
      if (od < rdv || (od == rdv && oi < riv)) { rdv = od; riv = oi; }
    }
    if (lane == 0) { swd[w] = rdv; swi[w] = riv; }
    __syncthreads();
    if (t == 0) {
      float wd = swd[0];
      int wi = swi[0];
#pragma unroll
      for (int q = 1; q < 8; ++q) {
        float qd = swd[q];
        int qi = swi[q];
        if (qd < wd || (qd == wd && qi < wi)) { wd = qd; wi = qi; }
      }
      swd[8] = wd;
      swi[8] = wi;
      idx_ws[i * KNB + it] = wi;
      dk_ws[i * KNB + it] = wd;
    }
    __syncthreads();
    const int wi = swi[8];
    if (t == (wi >> 5)) {
      // only the winning thread's local min changes: rescan its 32 slots
      mask |= 1u << (wi & 31);
      bd = 3.4e38f;
      bi = 0x7fffffff;
#pragma unroll
      for (int s = 0; s < 32; ++s) {
        if ((mask >> s) & 1u) continue;
        if (dl[s] < bd) { bd = dl[s]; bi = base + s; }
      }
    }
    // no barrier needed here: swi[8] is only rewritten after the next
    // iteration's first __syncthreads()
  }
}

// ---------------------------------------------------------------------------
// Kernel 3: fused edge featurization + GEMM1(416->128) + LayerNorm +
// GEMM2(128->128). One 128-thread (4-wave) block per 16 edge rows.
// Wave w owns output N-tiles 2w and 2w+1. Gathers go global->LDS via the
// CDNA5 async path (ASYNCcnt) and overlap the transcendental feature math.
// ---------------------------------------------------------------------------
__global__ __launch_bounds__(128) void edge_kernel(
    const float* __restrict__ X, const float* __restrict__ node_h,
    const float* __restrict__ f_node, const float* __restrict__ Wpos,
    const float* __restrict__ bpos, const float* __restrict__ ln_scale,
    const float* __restrict__ ln_bias, const float* __restrict__ bproj,
    const int* __restrict__ aatype, const int* __restrict__ ridx,
    const int* __restrict__ idx_ws, const float* __restrict__ dk_ws,
    const v2f* __restrict__ WeP, const v2f* __restrict__ WprojP,
    float* __restrict__ out) {
  __shared__ float ein[16][420];   // 16 x 416 e_in tile, padded stride
  __shared__ float e2[16][132];    // 16 x 128 e tile, padded stride
  __shared__ float red[16][8][2];
  __shared__ float rowstat[16][2];
  __shared__ int sj[16];

  const int tid = threadIdx.x;
  const int g0 = blockIdx.x * 16;
  const int lane = tid & 31;
  const int w = tid >> 5;
  const int half = lane >> 4;
  const int mrow = lane & 15;

  // ---- stage 0: publish neighbor indices (threads 0..15) ----
  int jr = 0;
  float dr = 0.f;
  if (tid < 16) {
    const int g = g0 + tid;
    const int i = g / KNB;
    const int k = g - i * KNB;
    jr = idx_ws[i * KNB + k];
    dr = dk_ws[i * KNB + k];
    sj[tid] = jr;
  }
  __syncthreads();

  // ---- stage 1a: launch gathers into e_in ----
  // cols: [0,128)=f_node[j], [160,288)=node_h[i], [288,416)=node_h[j]
  for (int m = tid; m < 1536; m += 128) {
    int r = m / 96;
    int q = m - r * 96;
    int reg = q >> 5;
    int c4 = q & 31;
    int g = g0 + r;
    int i = g / KNB;
    int j = sj[r];
    const float4* src;
    int dc;
    if (reg == 0) {
      src = (const float4*)&f_node[j * EDGEF]; dc = 0;
    } else if (reg == 1) {
      src = (const float4*)&node_h[i * NODEF]; dc = 160;
    } else {
      src = (const float4*)&node_h[j * NODEF]; dc = 288;
    }
#ifdef USE_ASYNC_GATHER
    __builtin_amdgcn_global_load_async_to_lds_b128(
        (v4i*)const_cast<float4*>(src + c4),
        AS3V4I(&ein[r][dc + 4 * c4]), 0, 0);
#else
    *(float4*)&ein[r][dc + 4 * c4] = src[c4];
#endif
  }

  // ---- stage 1b: per-row scalar features, overlapped with the gather ----
  if (tid < 16) {
    const int r = tid;
    const int g = g0 + r;
    const int i = g / KNB;
    const int j = jr;
    const float d = dr;

    float rb[16];
#pragma unroll
    for (int m = 0; m < 16; ++m) {
      float mu = 2.0f + (20.0f / 15.0f) * (float)m;  // linspace(2,22,16)
      float z = (d - mu) * (1.0f / 1.25f);           // sigma = 20/16
      rb[m] = expf(-z * z);
      ein[r][144 + m] = rb[m];
    }
    float invd = 1.0f / (d + 1e-6f);
    float geom[8];
    geom[0] = d * 0.1f;
    geom[1] = 1.0f / (1.0f + d);
    geom[2] = expf(-d);
    geom[3] = sinf(d);
    geom[4] = cosf(d);
    geom[5] = (X[3 * j + 0] - X[3 * i + 0]) * invd;
    geom[6] = (X[3 * j + 1] - X[3 * i + 1]) * invd;
    geom[7] = (X[3 * j + 2] - X[3 * i + 2]) * invd;
    int off = ridx[j] - ridx[i];
    int p = off + 8;
    p = p < 0 ? 0 : (p > 15 ? 15 : p);
    int aa = aatype[j];
    // e_pos = pos_in @ Wpos.T + bpos, exploiting one-hot sparsity.
    // pos_in layout: [pos_emb 0..15 | rbf 16..31 | geom 32..39 | seq 40..65]
#pragma unroll
    for (int o = 0; o < 16; ++o) {
      const float* wr = &Wpos[o * 66];
      float acc = bpos[o] + wr[p] + wr[40 + aa];
#pragma unroll
      for (int m = 0; m < 16; ++m) acc += rb[m] * wr[16 + m];
#pragma unroll
      for (int q = 0; q < 8; ++q) acc += geom[q] * wr[32 + q];
      ein[r][128 + o] = acc;
    }
  }
#ifdef USE_ASYNC_GATHER
  __builtin_amdgcn_s_wait_asynccnt(0);
#endif
  __syncthreads();

  // ---- stage 2: GEMM1 e = e_in @ We.T via v_wmma_f32_16x16x4_f32 ----
  {
    const int n0 = 2 * w;
    v8f acc0 = splat8(0.0f);
    v8f acc1 = splat8(0.0f);
    const v2f* bp0 = &WeP[(n0 * 104) * 32 + lane];
    const v2f* bp1 = &WeP[((n0 + 1) * 104) * 32 + lane];
    for (int cs = 0; cs < 104; ++cs) {
      v2f a = *reinterpret_cast<const v2f*>(&ein[mrow][4 * cs + 2 * half]);
      v2f b0 = bp0[cs * 32];
      v2f b1 = bp1[cs * 32];
      acc0 = wmma4(a, b0, acc0);
      acc1 = wmma4(a, b1, acc1);
    }
#pragma unroll
    for (int v = 0; v < 8; ++v) {
      e2[v + 8 * half][n0 * 16 + mrow] = acc0[v];
      e2[v + 8 * half][(n0 + 1) * 16 + mrow] = acc1[v];
    }
  }
  __syncthreads();

  // ---- stage 3: LayerNorm over 128 cols per row ----
  {
    int r = tid & 15, seg = tid >> 4;
    float s = 0.f, ss = 0.f;
#pragma unroll
    for (int c = 0; c < 16; ++c) {
      float v = e2[r][seg * 16 + c];
      s += v;
      ss += v * v;
    }
    red[r][seg][0] = s;
    red[r][seg][1] = ss;
  }
  __syncthreads();
  if (tid < 16) {
    float s = 0.f, ss = 0.f;
#pragma unroll
    for (int seg = 0; seg < 8; ++seg) {
      s += red[tid][seg][0];
      ss += red[tid][seg][1];
    }
    float mean = s * (1.0f / 128.0f);
    float var = ss * (1.0f / 128.0f) - mean * mean;  // biased, as jnp.var
    rowstat[tid][0] = mean;
    rowstat[tid][1] = rsqrtf(var + 1e-5f);
  }
  __syncthreads();
  {
    int r = tid & 15, seg = tid >> 4;
    float mean = rowstat[r][0], inv = rowstat[r][1];
#pragma unroll
    for (int c = 0; c < 16; ++c) {
      int col = seg * 16 + c;
      e2[r][col] = (e2[r][col] - mean) * inv * ln_scale[col] + ln_bias[col];
    }
  }
  __syncthreads();

  // ---- stage 4: GEMM2 out = e_norm @ Wproj.T + bproj ----
  {
    const int n0 = 2 * w;
    v8f acc0 = splat8(bproj[n0 * 16 + mrow]);
    v8f acc1 = splat8(bproj[(n0 + 1) * 16 + mrow]);
    const v2f* bp0 = &WprojP[(n0 * 32) * 32 + lane];
    const v2f* bp1 = &WprojP[((n0 + 1) * 32) * 32 + lane];
    for (int cs = 0; cs < 32; ++cs) {
      v2f a = *reinterpret_cast<const v2f*>(&e2[mrow][4 * cs + 2 * half]);
      acc0 = wmma4(a, bp0[cs * 32], acc0);
      acc1 = wmma4(a, bp1[cs * 32], acc1);
    }
#pragma unroll
    for (int v = 0; v < 8; ++v) {
      int row = g0 + v + 8 * half;
      out[row * EDGEF + n0 * 16 + mrow] = acc0[v];
      out[row * EDGEF + (n0 + 1) * 16 + mrow] = acc1[v];
    }
  }
}

// ---------------------------------------------------------------------------
extern "C" void kernel_launch(void* const* d_in, const int* in_sizes, int n_in,
                              void* d_out, int out_size, void* d_ws,
                              size_t ws_size, hipStream_t stream) {
  (void)in_sizes; (void)n_in; (void)out_size; (void)ws_size;
  const float* X      = (const float*)d_in[0];
  const float* node_h = (const float*)d_in[1];
  const float* f_node = (const float*)d_in[2];
  const float* Wpos   = (const float*)d_in[3];
  const float* bpos   = (const float*)d_in[4];
  const float* We     = (const float*)d_in[5];
  const float* ln_s   = (const float*)d_in[6];
  const float* ln_b   = (const float*)d_in[7];
  const float* Wproj  = (const float*)d_in[8];
  const float* bproj  = (const float*)d_in[9];
  const int* aatype   = (const int*)d_in[10];
  const int* ridx     = (const int*)d_in[11];
  float* out = (float*)d_out;

  // workspace layout (bytes): idx 1.5MB | Dk 1.5MB | WeP 208KB | WprojP 64KB
  char* ws = (char*)d_ws;
  int* idx_ws   = (int*)(ws);
  float* dk_ws  = (float*)(ws + 1572864);
  v2f* WeP      = (v2f*)(ws + 3145728);
  v2f* WprojP   = (v2f*)(ws + 3358720);

  pack_kernel<<<(26624 + 8192 + 255) / 256, 256, 0, stream>>>(We, Wproj, WeP,
                                                              WprojP);
  topk_kernel<<<LRES, 256, 0, stream>>>(X, idx_ws, dk_ws);
  edge_kernel<<<(LRES * KNB) / 16, 128, 0, stream>>>(
      X, node_h, f_node, Wpos, bpos, ln_s, ln_b, bproj, aatype, ridx, idx_ws,
      dk_ws, WeP, WprojP, out);
}